// CapsuleNet_72464688218163
// MI455X (gfx1250) — compile-verified
//
#include <hip/hip_runtime.h>
#include <math.h>

typedef float v2f __attribute__((ext_vector_type(2)));
typedef float v8f __attribute__((ext_vector_type(8)));
typedef __attribute__((address_space(3))) float lds_f32;

#define WMMA_F32(a, b, c) \
  __builtin_amdgcn_wmma_f32_16x16x4_f32(false, (a), false, (b), (short)0, (c), false, false)

// ---- problem constants ----
constexpr int BATCH  = 64;
constexpr int NNODES = 1296;
constexpr int NCAPS  = 8;
constexpr int NCLS   = 43;
constexpr int ODIM   = 16;
constexpr int C1_K   = 243;    // 3*9*9
constexpr int P_K    = 16384;  // 256*8*8

__device__ __forceinline__ uint32_t lds_offset(const float* p) {
  return (uint32_t)(uintptr_t)(lds_f32*)p;
}

// =====================================================================
// conv1: (64,3,32,32) -> (64,256,24,24), 9x9 valid. Implicit GEMM:
// M=36864, N=256, K=243. Block = 1 M-tile x full N. 8 waves x 2 N-tiles.
// A tile staged to LDS via global_load_async_to_lds_b32 (ASYNCcnt path).
// =====================================================================
__global__ __launch_bounds__(256) void conv1_wmma(
    const float* __restrict__ x, const float* __restrict__ w,
    const float* __restrict__ bias, float* __restrict__ h) {
  __shared__ float As[16][18];  // pitch 18: 8B-aligned float2, no bank conflicts
  const int t = threadIdx.x;
  const int lane = t & 31, wave = t >> 5;
  const int mtile = blockIdx.x;  // 2304
  const int hf = lane >> 4, l16 = lane & 15;

  // staging role: one element per thread per chunk
  const int srow = t & 15, skk = t >> 4;  // skk in 0..15
  const int sm  = mtile * 16 + srow;
  const int sb  = sm / 576;
  const int smr = sm % 576;
  const int soy = smr / 24, sox = smr % 24;
  const uint32_t sbase = (uint32_t)(((sb * 3) * 32 + soy) * 32 + sox);  // elem units
  const uint32_t ldso  = lds_offset(&As[srow][skk]);

  const int oc0 = (2 * wave) * 16 + l16;
  const int oc1 = (2 * wave + 1) * 16 + l16;
  const float* __restrict__ w0 = w + (size_t)oc0 * C1_K;
  const float* __restrict__ w1 = w + (size_t)oc1 * C1_K;
  v8f acc0 = {}, acc1 = {};

  // main chunks: k in [0,240), no guards anywhere
  for (int kb = 0; kb < 240; kb += 16) {
    {
      const int k  = kb + skk;
      const int c  = k / 81;
      const int r  = k % 81;
      const uint32_t voff = (sbase + (uint32_t)(c * 1024 + (r / 9) * 32 + (r % 9))) * 4u;
      asm volatile("global_load_async_to_lds_b32 %0, %1, %2"
                   :: "v"(ldso), "v"(voff), "s"(x) : "memory");
    }
    asm volatile("s_wait_asynccnt 0" ::: "memory");
    __syncthreads();
#pragma unroll
    for (int s = 0; s < 4; ++s) {
      const int kl = s * 4 + hf * 2;
      const v2f a = *(const v2f*)&As[l16][kl];
      const int kg = kb + kl;
      v2f b0, b1;
      b0[0] = w0[kg]; b0[1] = w0[kg + 1];
      b1[0] = w1[kg]; b1[1] = w1[kg + 1];
      acc0 = WMMA_F32(a, b0, acc0);
      acc1 = WMMA_F32(a, b1, acc1);
    }
    __syncthreads();
  }
  // tail chunk: k = 240..242 valid, one WMMA step
  {
    const int k = 240 + skk;
    if (k < C1_K) {
      const int c = k / 81;
      const int r = k % 81;
      const uint32_t voff = (sbase + (uint32_t)(c * 1024 + (r / 9) * 32 + (r % 9))) * 4u;
      asm volatile("global_load_async_to_lds_b32 %0, %1, %2"
                   :: "v"(ldso), "v"(voff), "s"(x) : "memory");
    } else {
      As[srow][skk] = 0.f;
    }
    asm volatile("s_wait_asynccnt 0" ::: "memory");
    __syncthreads();
    const int kl = hf * 2;
    const v2f a = *(const v2f*)&As[l16][kl];
    v2f b0, b1;
#pragma unroll
    for (int v = 0; v < 2; ++v) {
      const int k2 = 240 + kl + v;
      b0[v] = (k2 < C1_K) ? w0[k2] : 0.f;
      b1[v] = (k2 < C1_K) ? w1[k2] : 0.f;
    }
    acc0 = WMMA_F32(a, b0, acc0);
    acc1 = WMMA_F32(a, b1, acc1);
  }

  const float bia0 = bias[oc0], bia1 = bias[oc1];
#pragma unroll
  for (int r = 0; r < 8; ++r) {
    const int row = mtile * 16 + r + (hf << 3);
    const int bb = row / 576;
    const int rr = row % 576;
    float* hp = h + (size_t)bb * 147456 + rr;
    hp[(size_t)oc0 * 576] = fmaxf(acc0[r] + bia0, 0.f);
    hp[(size_t)oc1 * 576] = fmaxf(acc1[r] + bia1, 0.f);
  }
}

// =====================================================================
// prim conv: stride-2 8x8 -> implicit GEMM M=5184, N=128, K=16384.
// Block = 1 M-tile x full N. 8 waves x 1 N-tile. LDS-staged A.
// =====================================================================
__global__ __launch_bounds__(256) void prim_wmma(
    const float* __restrict__ h, const float* __restrict__ w,
    const float* __restrict__ bias, float* __restrict__ p) {
  __shared__ float As[16][34];  // KC=32, pitch 34
  const int t = threadIdx.x;
  const int lane = t & 31, wave = t >> 5;
  const int mtile = blockIdx.x;  // 324
  const int hf = lane >> 4, l16 = lane & 15;

  const int srow = t & 15, skk = t >> 4;
  const int sm  = mtile * 16 + srow;
  const int sb  = sm / 81;
  const int smr = sm % 81;
  const int soy = smr / 9, sox = smr % 9;
  // h[((sb*256+ic)*24 + 2*soy+ky)*24 + 2*sox+kx]
  const float* __restrict__ hrow = h + ((size_t)(sb * 256) * 24 + 2 * soy) * 24 + 2 * sox;

  const int oc = wave * 16 + l16;
  const float* __restrict__ wrow = w + (size_t)oc * P_K;
  v8f acc = {};

  for (int kb = 0; kb < P_K; kb += 32) {
    {
      const int k0 = kb + skk;
      const int k1 = k0 + 16;
      As[srow][skk] =
          hrow[(k0 >> 6) * 576 + ((k0 >> 3) & 7) * 24 + (k0 & 7)];
      As[srow][skk + 16] =
          hrow[(k1 >> 6) * 576 + ((k1 >> 3) & 7) * 24 + (k1 & 7)];
    }
    __syncthreads();
#pragma unroll
    for (int s = 0; s < 8; ++s) {
      const int kl = s * 4 + hf * 2;
      const v2f a = *(const v2f*)&As[l16][kl];
      const v2f b = *(const v2f*)(wrow + kb + kl);  // 8B aligned
      acc = WMMA_F32(a, b, acc);
    }
    __syncthreads();
  }

  const float bia = bias[oc];
#pragma unroll
  for (int r = 0; r < 8; ++r) {
    const int row = mtile * 16 + r + (hf << 3);
    const int bb = row / 81;
    const int sp = row % 81;
    p[(size_t)bb * 10368 + (size_t)oc * 81 + sp] = acc[r] + bia;
  }
}

// =====================================================================
// squash over the 8-dim capsule axis -> u[b][n][i]
// =====================================================================
__global__ void squash_u_k(const float* __restrict__ p, float* __restrict__ u) {
  const int idx = blockIdx.x * blockDim.x + threadIdx.x;  // b*1296+n
  if (idx >= BATCH * NNODES) return;
  const int b  = idx / NNODES;
  const int n  = idx % NNODES;
  const int hi = n / 81;
  const int sp = n % 81;
  float v[NCAPS];
  float sn = 0.f;
#pragma unroll
  for (int i = 0; i < NCAPS; ++i) {
    v[i] = p[(b * 128 + i * 16 + hi) * 81 + sp];
    sn += v[i] * v[i];
  }
  const float scale = sn / (1.f + sn) / sqrtf(sn);
#pragma unroll
  for (int i = 0; i < NCAPS; ++i) u[idx * NCAPS + i] = v[i] * scale;
}

__global__ void zero_f32(float* __restrict__ dst, int n) {
  const int i = blockIdx.x * blockDim.x + threadIdx.x;
  if (i < n) dst[i] = 0.f;
}

__global__ void softmax_c_k(const float* __restrict__ logits,
                            float* __restrict__ probs) {
  const int idx = blockIdx.x * blockDim.x + threadIdx.x;  // b*1296+n
  if (idx >= BATCH * NNODES) return;
  const float* row = logits + (size_t)idx * NCLS;
  float* out = probs + (size_t)idx * NCLS;
  float mx = row[0];
  for (int c = 1; c < NCLS; ++c) mx = fmaxf(mx, row[c]);
  float s = 0.f;
  for (int c = 0; c < NCLS; ++c) {
    const float e = expf(row[c] - mx);
    out[c] = e;
    s += e;
  }
  const float inv = 1.f / s;
  for (int c = 0; c < NCLS; ++c) out[c] *= inv;
}

// =====================================================================
// s[b][c][o] = sum_{n,i} (probs[b,n,c]*u[b,n,i]) * rw[n,c,i,o]
// One GEMM per class: M=64, K=1296*8, N=16.
// =====================================================================
__global__ __launch_bounds__(256) void routed_sum_wmma(
    const float* __restrict__ u, const float* __restrict__ probs,
    const float* __restrict__ rw, float* __restrict__ s) {
  const int lane = threadIdx.x & 31;
  const int wave = threadIdx.x >> 5;
  const int tile = blockIdx.x * 8 + wave;
  if (tile >= NCLS * 4) return;  // wave-uniform guard: EXEC stays all-1s
  const int c     = tile >> 2;
  const int mtile = tile & 3;
  const int hf  = lane >> 4;
  const int l16 = lane & 15;
  const int brow = mtile * 16 + l16;

  const float* __restrict__ up0 = u + (size_t)brow * NNODES * NCAPS;
  const float* __restrict__ pp0 = probs + (size_t)brow * NNODES * NCLS + c;
  const float* __restrict__ rw0 = rw + (size_t)c * NCAPS * ODIM + l16;

  v8f acc = {};
  for (int n = 0; n < NNODES; ++n) {
    const float pv = pp0[(size_t)n * NCLS];
    const float* up = up0 + (size_t)n * NCAPS;
    const float* rwp = rw0 + (size_t)n * (NCLS * NCAPS * ODIM);
#pragma unroll
    for (int st = 0; st < 2; ++st) {
      const int i0 = st * 4 + hf * 2;
      const v2f uv = *(const v2f*)(up + i0);  // 8B aligned
      v2f a, b;
      a[0] = pv * uv[0];
      a[1] = pv * uv[1];
      b[0] = rwp[(size_t)i0 * ODIM];
      b[1] = rwp[(size_t)(i0 + 1) * ODIM];
      acc = WMMA_F32(a, b, acc);
    }
  }
#pragma unroll
  for (int r = 0; r < 8; ++r) {
    const int bb = mtile * 16 + r + (hf << 3);
    s[((size_t)bb * NCLS + c) * ODIM + l16] = acc[r];
  }
}

__global__ void squash_out_k(const float* __restrict__ s,
                             float* __restrict__ outv) {
  const int idx = blockIdx.x * blockDim.x + threadIdx.x;  // b*43+c
  if (idx >= BATCH * NCLS) return;
  const float* row = s + (size_t)idx * ODIM;
  float sn = 0.f;
#pragma unroll
  for (int o = 0; o < ODIM; ++o) sn += row[o] * row[o];
  const float scale = sn / (1.f + sn) / sqrtf(sn);
  float* w = outv + (size_t)idx * ODIM;
#pragma unroll
  for (int o = 0; o < ODIM; ++o) w[o] = row[o] * scale;
}

// logits[b,n,c] += sum_{i,o} u[b,n,i] * outv[b,c,o] * rw[n,c,i,o]
__global__ void update_logits_k(const float* __restrict__ u,
                                const float* __restrict__ outv,
                                const float* __restrict__ rw,
                                float* __restrict__ logits) {
  const int idx = blockIdx.x * blockDim.x + threadIdx.x;
  if (idx >= BATCH * NNODES * NCLS) return;
  const int b = idx / (NNODES * NCLS);
  const int r = idx % (NNODES * NCLS);
  const int n = r / NCLS;
  const int c = r % NCLS;
  float ov[ODIM];
  const float* op = outv + ((size_t)b * NCLS + c) * ODIM;
#pragma unroll
  for (int o = 0; o < ODIM; ++o) ov[o] = op[o];
  float acc = 0.f;
#pragma unroll
  for (int i = 0; i < NCAPS; ++i) {
    const float ui = u[((size_t)b * NNODES + n) * NCAPS + i];
    const float* rwp = rw + (((size_t)n * NCLS + c) * NCAPS + i) * ODIM;
    float d = 0.f;
#pragma unroll
    for (int o = 0; o < ODIM; ++o) d += ov[o] * rwp[o];
    acc += ui * d;
  }
  logits[((size_t)b * NNODES + n) * NCLS + c] += acc;
}

__global__ void scores_k(const float* __restrict__ outv, float* __restrict__ out) {
  const int idx = blockIdx.x * blockDim.x + threadIdx.x;  // b*43+c
  if (idx >= BATCH * NCLS) return;
  const float* row = outv + (size_t)idx * ODIM;
  float sn = 0.f;
#pragma unroll
  for (int o = 0; o < ODIM; ++o) sn += row[o] * row[o];
  out[idx] = sqrtf(sn);
}

// =====================================================================
extern "C" void kernel_launch(void* const* d_in, const int* in_sizes, int n_in,
                              void* d_out, int out_size, void* d_ws, size_t ws_size,
                              hipStream_t stream) {
  const float* x       = (const float*)d_in[0];
  const float* conv1_w = (const float*)d_in[1];
  const float* conv1_b = (const float*)d_in[2];
  const float* prim_w  = (const float*)d_in[3];
  const float* prim_b  = (const float*)d_in[4];
  const float* route_w = (const float*)d_in[5];
  float* out = (float*)d_out;

  float* ws     = (float*)d_ws;
  float* h      = ws;                         // 9,437,184
  float* p      = h + (size_t)9437184;        //   663,552
  float* u      = p + (size_t)663552;         //   663,552
  float* logits = u + (size_t)663552;         // 3,571,712
  float* probs  = logits + (size_t)3571712;   // 3,571,712
  float* s      = probs + (size_t)3571712;    //    44,032
  float* outv   = s + (size_t)44032;          //    44,032

  conv1_wmma<<<2304, 256, 0, stream>>>(x, conv1_w, conv1_b, h);
  prim_wmma<<<324, 256, 0, stream>>>(h, prim_w, prim_b, p);
  squash_u_k<<<(BATCH * NNODES + 255) / 256, 256, 0, stream>>>(p, u);
  zero_f32<<<(BATCH * NNODES * NCLS + 255) / 256, 256, 0, stream>>>(
      logits, BATCH * NNODES * NCLS);

  for (int it = 0; it < 3; ++it) {
    softmax_c_k<<<(BATCH * NNODES + 255) / 256, 256, 0, stream>>>(logits, probs);
    routed_sum_wmma<<<(NCLS * 4 + 7) / 8, 256, 0, stream>>>(u, probs, route_w, s);
    squash_out_k<<<(BATCH * NCLS + 255) / 256, 256, 0, stream>>>(s, outv);
    if (it != 2) {
      update_logits_k<<<(BATCH * NNODES * NCLS + 255) / 256, 256, 0, stream>>>(
          u, outv, route_w, logits);
    }
  }
  scores_k<<<(BATCH * NCLS + 255) / 256, 256, 0, stream>>>(outv, out);
}